// TimeStepAwareModalityFusion_29540785061969
// MI455X (gfx1250) — compile-verified
//
#include <hip/hip_runtime.h>
#include <math.h>

typedef __attribute__((ext_vector_type(2))) float v2f;
typedef __attribute__((ext_vector_type(8))) float v8f;

#define C_DIM 256
#define D_DIM 256
#define N_TOK 4096
#define SCALE 0.0625f   // 1/sqrt(256)
#define EPSV  1e-5f

// ---- workspace layout (floats) ----
#define WS_GP    0        // [4*256] g_c * p_c per modality
#define WS_SGP   1024     // [4] sum_c g_c p_c
#define WS_SBP   1028     // [4] sum_c be_c p_c
#define WS_LOG   1032     // [16384] logits
#define WS_MU    17416    // [16384] token means
#define WS_RSTD  33800    // [16384] token rstd
#define WS_ALPHA 50184    // [16384] attn_j * rstd_j
#define WS_WSUM  66568    // [4] sum_j attn_j
#define WS_AMU   66572    // [4] sum_j alpha_j mu_j
#define WS_U     66576    // [4*256] reduced LN-weighted channel vector
#define WS_YVEC  67600    // [256] final channel vector

// One 16-column tile of y = x(1xK) @ W(KxN), W element (k,n) at w[k*sk + n*sn].
// A matrix: only row M=0 live (holds x); rows 1..15 zeroed via a hoisted
// multiplicative lane mask (no EXEC divergence -> WMMA-legal, branch-free).
// Result row 0 lands in acc[0] of lanes 0..15.
__device__ inline v8f gemv_tile_f32(const float* x, const float* w,
                                    int K, long sk, long sn, int n0, v8f acc) {
    const int lane = threadIdx.x & 31;
    const int half = lane >> 4;            // 0: K pair (k0,k0+1)  1: (k0+2,k0+3)
    const int l16  = lane & 15;
    const float amask = (l16 == 0) ? 1.0f : 0.0f;   // hoisted, branch-free
    const float* wp = w + (long)(n0 + l16) * sn;
    for (int k0 = 0; k0 < K; k0 += 4) {
        const int kb = k0 + 2 * half;
        const float x0 = x[kb];            // unconditional (uniform addr per half)
        const float x1 = x[kb + 1];
        v2f a;
        a.x = x0 * amask;                  // A[M=l16][kb]     (only M=0 live)
        a.y = x1 * amask;                  // A[M=l16][kb+1]
        v2f b;
        b.x = wp[(long)kb * sk];           // B[kb][n0+l16]
        b.y = wp[(long)(kb + 1) * sk];     // B[kb+1][n0+l16]
        acc = __builtin_amdgcn_wmma_f32_16x16x4_f32(
            /*neg_a=*/false, a, /*neg_b=*/false, b,
            /*c_mod=*/(short)0, acc, /*reuse_a=*/false, /*reuse_b=*/false);
    }
    return acc;
}

__device__ inline float block_reduce_sum_512(float v, float* red) {
    const int t = threadIdx.x;
    red[t] = v; __syncthreads();
    for (int o = 256; o > 0; o >>= 1) {
        if (t < o) red[t] += red[t + o];
        __syncthreads();
    }
    float r = red[0]; __syncthreads();
    return r;
}

// ---------------- Kernel 1: q = t@wt+bt ; p_mod = wk_mod @ q ; g*p and sums --
__global__ __launch_bounds__(512) void prep_kernel(
    const float* tvec, const float* wt, const float* bt,
    const float* wk_x, const float* wk_h, const float* wk_s, const float* wk_m,
    const float* g_x, const float* be_x, const float* g_h, const float* be_h,
    const float* g_s, const float* be_s, const float* g_m, const float* be_m,
    float* gp, float* sgp, float* sbp)
{
    __shared__ float q_s[D_DIM];
    __shared__ float p_s[C_DIM];
    __shared__ float red[512];
    const int tid  = threadIdx.x;
    const int wave = tid >> 5;
    const int lane = tid & 31;

    // q = t @ wt + bt : K=C(256), N=D(256) -> 16 tiles over 16 waves
    {
        v8f acc = {};
        acc = gemv_tile_f32(tvec, wt, C_DIM, (long)D_DIM, 1L, wave * 16, acc);
        if (lane < 16) q_s[wave * 16 + lane] = acc[0] + bt[wave * 16 + lane];
    }
    __syncthreads();

    const float* wks[4] = {wk_x, wk_h, wk_s, wk_m};
    const float* gs[4]  = {g_x,  g_h,  g_s,  g_m};
    const float* bes[4] = {be_x, be_h, be_s, be_m};

    for (int mod = 0; mod < 4; ++mod) {
        // p[c] = sum_d wk[c*D+d]*q[d]  => W(k=d,n=c): sk=1, sn=D
        v8f acc = {};
        acc = gemv_tile_f32(q_s, wks[mod], D_DIM, 1L, (long)D_DIM, wave * 16, acc);
        if (lane < 16) p_s[wave * 16 + lane] = acc[0];
        __syncthreads();

        float gv = 0.0f, bv = 0.0f;
        if (tid < C_DIM) {
            const float p = p_s[tid];
            gv = gs[mod][tid] * p;
            bv = bes[mod][tid] * p;
            gp[mod * C_DIM + tid] = gv;
        }
        const float sg = block_reduce_sum_512(gv, red);
        const float sb = block_reduce_sum_512(bv, red);
        if (tid == 0) { sgp[mod] = sg; sbp[mod] = sb; }
        __syncthreads();
    }
}

// ---------- Kernel 2: per-token LN stats + logit (streams all 4 tensors) ----
// 256 blocks: each handles 64 tokens of one modality with a 4-way channel
// split (grp = channel group, tl = token lane). Coalesced: each wave reads
// 32 consecutive tokens per channel.
__global__ __launch_bounds__(256) void pass1_kernel(
    const float* x, const float* h, const float* s, const float* m,
    const float* gp, const float* sgp, const float* sbp,
    float* logits, float* mus, float* rstds)
{
    __shared__ float gpm_s[C_DIM];
    __shared__ float sh1[4][64];
    __shared__ float sh2[4][64];
    __shared__ float shd[4][64];
    const int bid = blockIdx.x;                 // 256 blocks
    const int mod = bid >> 6;                   // 64 blocks per modality
    const int j0  = (bid & 63) << 6;            // token base (64 tokens/block)
    const int tl  = threadIdx.x & 63;           // token lane
    const int grp = threadIdx.x >> 6;           // channel group (0..3)
    const float* aptr[4] = {x, h, s, m};
    const float* a = aptr[mod];

    gpm_s[threadIdx.x] = gp[mod * C_DIM + threadIdx.x];
    __syncthreads();

    const int j  = j0 + tl;
    const int c0 = grp << 6;                    // 64 channels per group
    float s1 = 0.0f, s2 = 0.0f, dp = 0.0f;
#pragma unroll 4
    for (int cc = 0; cc < 64; ++cc) {
        const int c = c0 + cc;
        const float v = a[(long)c * N_TOK + j];
        s1 += v;
        s2 += v * v;
        dp += v * gpm_s[c];
    }
    sh1[grp][tl] = s1; sh2[grp][tl] = s2; shd[grp][tl] = dp;
    __syncthreads();

    if (threadIdx.x < 64) {
        const int jj = j0 + threadIdx.x;
        const float t1 = sh1[0][threadIdx.x] + sh1[1][threadIdx.x] +
                         sh1[2][threadIdx.x] + sh1[3][threadIdx.x];
        const float t2 = sh2[0][threadIdx.x] + sh2[1][threadIdx.x] +
                         sh2[2][threadIdx.x] + sh2[3][threadIdx.x];
        const float td = shd[0][threadIdx.x] + shd[1][threadIdx.x] +
                         shd[2][threadIdx.x] + shd[3][threadIdx.x];
        const float mu   = t1 * (1.0f / C_DIM);
        const float var  = t2 * (1.0f / C_DIM) - mu * mu;
        const float rstd = rsqrtf(var + EPSV);
        const float lg   = SCALE * (rstd * (td - mu * sgp[mod]) + sbp[mod]);
        const int idx = mod * N_TOK + jj;
        logits[idx] = lg;
        mus[idx]    = mu;
        rstds[idx]  = rstd;
    }
}

// ----- Kernel 3: softmax over 16384 logits + per-mod scalar reductions ------
__global__ __launch_bounds__(1024) void softmax_kernel(
    const float* logits, const float* mus, const float* rstds,
    float* alpha, float* wsum, float* amu)
{
    __shared__ float red[1024];
    const int t = threadIdx.x;
    float l[16];
    float lmax = -3.4e38f;
#pragma unroll
    for (int i = 0; i < 16; ++i) { l[i] = logits[t * 16 + i]; lmax = fmaxf(lmax, l[i]); }
    red[t] = lmax; __syncthreads();
    for (int o = 512; o > 0; o >>= 1) { if (t < o) red[t] = fmaxf(red[t], red[t + o]); __syncthreads(); }
    const float gmax = red[0]; __syncthreads();

    float ls = 0.0f;
#pragma unroll
    for (int i = 0; i < 16; ++i) { l[i] = __expf(l[i] - gmax); ls += l[i]; }
    red[t] = ls; __syncthreads();
    for (int o = 512; o > 0; o >>= 1) { if (t < o) red[t] += red[t + o]; __syncthreads(); }
    const float inv = 1.0f / red[0]; __syncthreads();

    const int mod = t >> 8;   // contiguous 16-chunks never cross mod boundary
    float wp = 0.0f, ap = 0.0f;
#pragma unroll
    for (int i = 0; i < 16; ++i) {
        const int j  = t * 16 + i;
        const float w  = l[i] * inv;
        const float al = w * rstds[j];
        alpha[j] = al;
        wp += w;
        ap += al * mus[j];
    }
    for (int md = 0; md < 4; ++md) {
        red[t] = (mod == md) ? wp : 0.0f; __syncthreads();
        for (int o = 512; o > 0; o >>= 1) { if (t < o) red[t] += red[t + o]; __syncthreads(); }
        if (t == 0) wsum[md] = red[0];
        __syncthreads();
        red[t] = (mod == md) ? ap : 0.0f; __syncthreads();
        for (int o = 512; o > 0; o >>= 1) { if (t < o) red[t] += red[t + o]; __syncthreads(); }
        if (t == 0) amu[md] = red[0];
        __syncthreads();
    }
}

// -- Kernel 4: u[mod][c] = g_c*(Σ_j α_j a_jc − Amu) + be_c*Wsum (2nd stream) --
__global__ __launch_bounds__(256) void pass2_kernel(
    const float* x, const float* h, const float* s, const float* m,
    const float* alpha, const float* wsum, const float* amu,
    const float* g_x, const float* be_x, const float* g_h, const float* be_h,
    const float* g_s, const float* be_s, const float* g_m, const float* be_m,
    float* u)
{
    __shared__ float red[256];
    const int bid = blockIdx.x;                 // 1024 blocks
    const int mod = bid >> 8;
    const int c   = bid & 255;
    const float* aptr[4] = {x, h, s, m};
    const float* gs[4]   = {g_x,  g_h,  g_s,  g_m};
    const float* bes[4]  = {be_x, be_h, be_s, be_m};

    const float* a  = aptr[mod] + (long)c * N_TOK;
    const float* al = alpha + mod * N_TOK;
    float acc = 0.0f;
#pragma unroll 4
    for (int j = threadIdx.x; j < N_TOK; j += 256) acc += al[j] * a[j];
    red[threadIdx.x] = acc; __syncthreads();
    for (int o = 128; o > 0; o >>= 1) {
        if (threadIdx.x < o) red[threadIdx.x] += red[threadIdx.x + o];
        __syncthreads();
    }
    if (threadIdx.x == 0) {
        const float T = red[0];
        u[mod * C_DIM + c] = gs[mod][c] * (T - amu[mod]) + bes[mod][c] * wsum[mod];
    }
}

// ---- Kernel 5: wv projection, LN, GEGLU, FF2, out-proj (all WMMA GEMVs) ----
__global__ __launch_bounds__(512) void head_kernel(
    const float* u,
    const float* wv_x, const float* wv_h, const float* wv_s, const float* wv_m,
    const float* g_res, const float* be_res,
    const float* w_geglu, const float* b_geglu,
    const float* w_ff2, const float* b_ff2,
    const float* w_out, const float* b_out,
    float* yvec)
{
    __shared__ float ovec[D_DIM];
    __shared__ float rr[D_DIM];
    __shared__ float ggl[2 * D_DIM];
    __shared__ float hh[D_DIM];
    __shared__ float ffv[D_DIM];
    __shared__ float red[512];
    const int tid  = threadIdx.x;
    const int wave = tid >> 5;
    const int lane = tid & 31;
    const float* wv[4] = {wv_x, wv_h, wv_s, wv_m};

    // out_vec = Σ_mod u_mod @ wv_mod  (accumulate across modalities in WMMA C)
    {
        v8f acc = {};
        for (int mod = 0; mod < 4; ++mod)
            acc = gemv_tile_f32(u + mod * C_DIM, wv[mod], C_DIM, (long)D_DIM, 1L,
                                wave * 16, acc);
        if (lane < 16) ovec[wave * 16 + lane] = acc[0];
    }
    __syncthreads();

    // LayerNorm over D=256
    float v = (tid < D_DIM) ? ovec[tid] : 0.0f;
    const float mu = block_reduce_sum_512(v, red) * (1.0f / D_DIM);
    const float dv = (tid < D_DIM) ? (ovec[tid] - mu) : 0.0f;
    const float var = block_reduce_sum_512(dv * dv, red) * (1.0f / D_DIM);
    const float rstd = rsqrtf(var + EPSV);
    if (tid < D_DIM) rr[tid] = (ovec[tid] - mu) * rstd * g_res[tid] + be_res[tid];
    __syncthreads();

    // GEGLU input: gg = r @ w_geglu + b_geglu (N=512 -> 32 tiles, 2 per wave)
    for (int tt = wave; tt < 32; tt += 16) {
        v8f acc = {};
        acc = gemv_tile_f32(rr, w_geglu, D_DIM, (long)(2 * D_DIM), 1L, tt * 16, acc);
        if (lane < 16) ggl[tt * 16 + lane] = acc[0] + b_geglu[tt * 16 + lane];
    }
    __syncthreads();

    // h = u_part * gelu_exact(gate)
    if (tid < D_DIM) {
        const float g  = ggl[D_DIM + tid];
        const float ge = 0.5f * g * (1.0f + erff(g * 0.70710678118654752f));
        hh[tid] = ggl[tid] * ge;
    }
    __syncthreads();

    // ff = h @ w_ff2 + b_ff2
    {
        v8f acc = {};
        acc = gemv_tile_f32(hh, w_ff2, D_DIM, (long)D_DIM, 1L, wave * 16, acc);
        if (lane < 16) ffv[wave * 16 + lane] = acc[0] + b_ff2[wave * 16 + lane];
    }
    __syncthreads();

    // y = ff @ w_out + b_out  -> global yvec
    {
        v8f acc = {};
        acc = gemv_tile_f32(ffv, w_out, D_DIM, (long)C_DIM, 1L, wave * 16, acc);
        if (lane < 16) yvec[wave * 16 + lane] = acc[0] + b_out[wave * 16 + lane];
    }
}

// ------------- Kernel 6: broadcast yvec over the 64x64 spatial grid ---------
__global__ __launch_bounds__(1024) void bcast_kernel(const float* yvec, float* out)
{
    const int c = blockIdx.x;
    const float v = yvec[c];
    float* o = out + (long)c * N_TOK;
    for (int j = threadIdx.x; j < N_TOK; j += 1024) o[j] = v;
}

extern "C" void kernel_launch(void* const* d_in, const int* in_sizes, int n_in,
                              void* d_out, int out_size, void* d_ws, size_t ws_size,
                              hipStream_t stream) {
    const float* x       = (const float*)d_in[0];
    const float* h       = (const float*)d_in[1];
    const float* s       = (const float*)d_in[2];
    const float* m       = (const float*)d_in[3];
    const float* tvec    = (const float*)d_in[4];
    const float* wt      = (const float*)d_in[5];
    const float* bt      = (const float*)d_in[6];
    const float* wk_x    = (const float*)d_in[7];
    const float* wv_x    = (const float*)d_in[8];
    const float* g_x     = (const float*)d_in[9];
    const float* be_x    = (const float*)d_in[10];
    const float* wk_h    = (const float*)d_in[11];
    const float* wv_h    = (const float*)d_in[12];
    const float* g_h     = (const float*)d_in[13];
    const float* be_h    = (const float*)d_in[14];
    const float* wk_s    = (const float*)d_in[15];
    const float* wv_s    = (const float*)d_in[16];
    const float* g_s     = (const float*)d_in[17];
    const float* be_s    = (const float*)d_in[18];
    const float* wk_m    = (const float*)d_in[19];
    const float* wv_m    = (const float*)d_in[20];
    const float* g_m     = (const float*)d_in[21];
    const float* be_m    = (const float*)d_in[22];
    const float* w_geglu = (const float*)d_in[23];
    const float* b_geglu = (const float*)d_in[24];
    const float* w_ff2   = (const float*)d_in[25];
    const float* b_ff2   = (const float*)d_in[26];
    const float* g_res   = (const float*)d_in[27];
    const float* be_res  = (const float*)d_in[28];
    const float* w_out   = (const float*)d_in[29];
    const float* b_out   = (const float*)d_in[30];
    float* out = (float*)d_out;
    float* ws  = (float*)d_ws;

    float* gp     = ws + WS_GP;
    float* sgp    = ws + WS_SGP;
    float* sbp    = ws + WS_SBP;
    float* logits = ws + WS_LOG;
    float* mus    = ws + WS_MU;
    float* rstds  = ws + WS_RSTD;
    float* alpha  = ws + WS_ALPHA;
    float* wsum   = ws + WS_WSUM;
    float* amu    = ws + WS_AMU;
    float* u      = ws + WS_U;
    float* yvec   = ws + WS_YVEC;

    prep_kernel<<<1, 512, 0, stream>>>(tvec, wt, bt,
                                       wk_x, wk_h, wk_s, wk_m,
                                       g_x, be_x, g_h, be_h, g_s, be_s, g_m, be_m,
                                       gp, sgp, sbp);
    pass1_kernel<<<256, 256, 0, stream>>>(x, h, s, m, gp, sgp, sbp,
                                          logits, mus, rstds);
    softmax_kernel<<<1, 1024, 0, stream>>>(logits, mus, rstds, alpha, wsum, amu);
    pass2_kernel<<<1024, 256, 0, stream>>>(x, h, s, m, alpha, wsum, amu,
                                           g_x, be_x, g_h, be_h, g_s, be_s, g_m, be_m,
                                           u);
    head_kernel<<<1, 512, 0, stream>>>(u, wv_x, wv_h, wv_s, wv_m,
                                       g_res, be_res, w_geglu, b_geglu,
                                       w_ff2, b_ff2, w_out, b_out, yvec);
    bcast_kernel<<<256, 1024, 0, stream>>>(yvec, out);
}